// GraphSAGE_78374563217856
// MI455X (gfx1250) — compile-verified
//
#include <hip/hip_runtime.h>
#include <math.h>

// ---------------------------------------------------------------------------
// GraphSAGE on MI455X (gfx1250): L2-resident scatter/gather + fp32 WMMA GEMMs
// with LDS-staged weights (conflict-free 136-float row stride) and
// register-staged fragments (clause-friendly loads, back-to-back WMMAs)
// ---------------------------------------------------------------------------
#define GN   50000      // nodes
#define GE   800000     // edges
#define GD   128        // feature dim
#define DOUT 2

typedef __attribute__((ext_vector_type(2))) float v2f;
typedef __attribute__((ext_vector_type(8))) float v8f;

#define AG_SUM  0
#define AG_MAX  1
#define AG_MEAN 2

// ---------------------------------------------------------------------------
// float atomic-max via signed/unsigned int atomics (classic monotone trick)
// ---------------------------------------------------------------------------
__device__ __forceinline__ void atomicMaxF(float* addr, float v) {
    if (v >= 0.0f)
        atomicMax(reinterpret_cast<int*>(addr), __float_as_int(v));
    else
        atomicMin(reinterpret_cast<unsigned int*>(addr), __float_as_uint(v));
}

// ---------------------------------------------------------------------------
// init agg (0 for sum/mean, -inf for max) and counts (0)
// ---------------------------------------------------------------------------
__global__ void sage_init(float* __restrict__ agg, float* __restrict__ cnt,
                          float fill) {
    long long i = (long long)blockIdx.x * blockDim.x + threadIdx.x;
    const long long tot = (long long)GN * GD;
    if (i < tot) agg[i] = fill;
    else if (i < tot + GN) cnt[i - tot] = 0.0f;
}

// ---------------------------------------------------------------------------
// per-edge scatter: 32 lanes per edge, float4 (16B) per lane
// gather h[src] (L2-resident) -> atomic aggregate into agg[dst]
// ---------------------------------------------------------------------------
__global__ void sage_scatter(const float* __restrict__ h,
                             const int* __restrict__ ei,
                             float* __restrict__ agg,
                             float* __restrict__ cnt, int aggr) {
    long long t = (long long)blockIdx.x * blockDim.x + threadIdx.x;
    int e    = (int)(t >> 5);
    int lane = (int)(t & 31);
    if (e >= GE) return;
    int src = ei[e];
    int dst = ei[GE + e];
    const float4 v =
        reinterpret_cast<const float4*>(h + (size_t)src * GD)[lane];
    float* ap = agg + (size_t)dst * GD + lane * 4;
    if (aggr == AG_MAX) {
        atomicMaxF(ap + 0, v.x);
        atomicMaxF(ap + 1, v.y);
        atomicMaxF(ap + 2, v.z);
        atomicMaxF(ap + 3, v.w);
    } else {
        atomicAdd(ap + 0, v.x);
        atomicAdd(ap + 1, v.y);
        atomicAdd(ap + 2, v.z);
        atomicAdd(ap + 3, v.w);
        if (aggr == AG_MEAN && lane == 0) atomicAdd(cnt + dst, 1.0f);
    }
}

// ---------------------------------------------------------------------------
// finalize: max -> replace -inf (no in-edges) with 0; mean -> divide by count
// ---------------------------------------------------------------------------
__global__ void sage_finalize(float* __restrict__ agg,
                              const float* __restrict__ cnt, int aggr) {
    long long i = (long long)blockIdx.x * blockDim.x + threadIdx.x;
    if (i >= (long long)GN * GD) return;
    float v = agg[i];
    if (aggr == AG_MAX) {
        if (!(v >= -3.0e38f)) v = 0.0f;   // catches -inf / NaN
        agg[i] = v;
    } else {  // mean
        float c = cnt[i >> 7];
        agg[i] = v / fmaxf(c, 1.0f);
    }
}

// ---------------------------------------------------------------------------
// fused dense step:  out = relu( A*W + bias [+ A2*W2] )
// One wave = one 16x16 tile via V_WMMA_F32_16X16X4_F32 (fp32 in/acc, K=4).
//   A fragment (16x4): lane m=l&15; vgpr pair holds K = 2*(l>>4) .. +1
//   B fragment (4x16): lane n=l&15; same K split (mirrors A)
//   C/D (16x16):       lane n=l&15; vgpr v holds row M = v + 8*(l>>4)
// Weights staged into LDS in 32-row K-panels (row stride 136 floats -> the
// lane<16 / lane>=16 bank windows differ by 16 banks, conflict-free
// ds_load_2addr_b32 offset1:136). Staging loads go through 4 distinct
// temporaries so the global_load_b128s clause and overlap before the
// ds_store_b128 burst. Per panel, ALL fragments are register-staged
// (clauseable global/ds load bursts), then the 8(+8) WMMAs issue
// back-to-back on two independent accumulator chains.
// block = 256 threads = 8 waves -> all 128 output cols of a 16-row stripe.
// grid.x = GN/16 = 3125 (exact)
// ---------------------------------------------------------------------------
#define SW 136          // LDS row stride in floats (2*SW % 64 == 16)

template <bool DUAL>
__global__ void sage_gemm_wmma(const float* __restrict__ A,
                               const float* __restrict__ W,
                               const float* __restrict__ bias,
                               const float* __restrict__ A2,
                               const float* __restrict__ W2,
                               float* __restrict__ out) {
    __shared__ float sW [32 * SW];
    __shared__ float sW2[DUAL ? 32 * SW : 1];

    const int lane = threadIdx.x & 31;
    const int wave = threadIdx.x >> 5;
    const int m0   = blockIdx.x * 16;
    const int n0   = wave * 16;
    const int mr   = lane & 15;
    const int hi   = lane >> 4;   // 0 or 1
    const int ks   = hi * 2;      // K sub-offset within a group of 4

    // staging map: thread -> (row 0..31, col base 0..112 step 16)
    const int lr = threadIdx.x >> 3;
    const int lc = (threadIdx.x & 7) * 16;

    v8f acc = {}, acc2 = {};
    const float* arow  = A + (size_t)(m0 + mr) * GD;
    const float* a2row = DUAL ? (A2 + (size_t)(m0 + mr) * GD) : A;
    const int bcol = n0 + mr;

    for (int kb = 0; kb < GD; kb += 32) {
        // ---- stage 32xGD weight panel(s) into LDS ----
        // load all float4s into distinct temps first so the b128 global
        // loads clause; single wait; then burst the ds_store_b128s.
        {
            const float4* src =
                reinterpret_cast<const float4*>(W + (size_t)(kb + lr) * GD + lc);
            float4 t0 = src[0], t1 = src[1], t2 = src[2], t3 = src[3];
            float4 u0, u1, u2, u3;
            if (DUAL) {
                const float4* src2 = reinterpret_cast<const float4*>(
                    W2 + (size_t)(kb + lr) * GD + lc);
                u0 = src2[0]; u1 = src2[1]; u2 = src2[2]; u3 = src2[3];
            }
            float4* dst = reinterpret_cast<float4*>(sW + lr * SW + lc);
            dst[0] = t0; dst[1] = t1; dst[2] = t2; dst[3] = t3;
            if (DUAL) {
                float4* dst2 = reinterpret_cast<float4*>(sW2 + lr * SW + lc);
                dst2[0] = u0; dst2[1] = u1; dst2[2] = u2; dst2[3] = u3;
            }
        }
        __syncthreads();

        // ---- register-stage all fragments for this panel ----
        v2f aT[8], bT[8], a2T[8], b2T[8];
#pragma unroll
        for (int kk = 0; kk < 8; ++kk) {
            const int k = kk * 4;
            const float2 af =
                *reinterpret_cast<const float2*>(arow + kb + k + ks);
            aT[kk].x = af.x; aT[kk].y = af.y;
            bT[kk].x = sW[(k + ks) * SW + bcol];
            bT[kk].y = sW[(k + ks + 1) * SW + bcol];
            if (DUAL) {
                const float2 af2 =
                    *reinterpret_cast<const float2*>(a2row + kb + k + ks);
                a2T[kk].x = af2.x; a2T[kk].y = af2.y;
                b2T[kk].x = sW2[(k + ks) * SW + bcol];
                b2T[kk].y = sW2[(k + ks + 1) * SW + bcol];
            }
        }

        // ---- back-to-back WMMA chain (two independent accumulators) ----
#pragma unroll
        for (int kk = 0; kk < 8; ++kk) {
            acc = __builtin_amdgcn_wmma_f32_16x16x4_f32(
                false, aT[kk], false, bT[kk], (short)0, acc, false, false);
            if (DUAL)
                acc2 = __builtin_amdgcn_wmma_f32_16x16x4_f32(
                    false, a2T[kk], false, b2T[kk], (short)0, acc2, false, false);
        }
        __syncthreads();
    }

    const float bv = bias[bcol];
#pragma unroll
    for (int v = 0; v < 8; ++v) {
        int row = m0 + v + hi * 8;
        float r = acc[v] + bv;
        if (DUAL) r += acc2[v];
        r = fmaxf(r, 0.0f);
        out[(size_t)row * GD + bcol] = r;
    }
}

// ---------------------------------------------------------------------------
// final skinny projection 128 -> 2: one wave per node, coalesced float4 row
// fetch, wave32 shuffle-xor reduction, lane 0 writes.
// ---------------------------------------------------------------------------
__global__ void sage_head(const float* __restrict__ h,
                          const float* __restrict__ w2,
                          const float* __restrict__ b2,
                          float* __restrict__ out) {
    int wid  = (int)(((long long)blockIdx.x * blockDim.x + threadIdx.x) >> 5);
    int lane = threadIdx.x & 31;
    if (wid >= GN) return;

    const float4 v =
        reinterpret_cast<const float4*>(h + (size_t)wid * GD)[lane];
    const int kb = lane * 4;
    float a0 = v.x * w2[(kb + 0) * DOUT]     + v.y * w2[(kb + 1) * DOUT]
             + v.z * w2[(kb + 2) * DOUT]     + v.w * w2[(kb + 3) * DOUT];
    float a1 = v.x * w2[(kb + 0) * DOUT + 1] + v.y * w2[(kb + 1) * DOUT + 1]
             + v.z * w2[(kb + 2) * DOUT + 1] + v.w * w2[(kb + 3) * DOUT + 1];

#pragma unroll
    for (int off = 16; off > 0; off >>= 1) {
        a0 += __shfl_xor(a0, off, 32);
        a1 += __shfl_xor(a1, off, 32);
    }
    if (lane == 0) {
        out[wid * DOUT + 0] = a0 + b2[0];
        out[wid * DOUT + 1] = a1 + b2[1];
    }
}

// ---------------------------------------------------------------------------
// launch
// ---------------------------------------------------------------------------
extern "C" void kernel_launch(void* const* d_in, const int* in_sizes, int n_in,
                              void* d_out, int out_size, void* d_ws,
                              size_t ws_size, hipStream_t stream) {
    const float* x   = (const float*)d_in[0];   // [N,128]
    const int*   ei  = (const int*)d_in[1];     // [2,E]
    const float* w_l = (const float*)d_in[2];   // [6,128,128]
    const float* b_l = (const float*)d_in[3];   // [6,128]
    const float* w_r = (const float*)d_in[4];   // [6,128,128]
    const float* w1  = (const float*)d_in[5];   // [128,128]
    const float* b1  = (const float*)d_in[6];   // [128]
    const float* w2  = (const float*)d_in[7];   // [128,2]
    const float* b2  = (const float*)d_in[8];   // [2]
    float*       out = (float*)d_out;

    // workspace layout (floats)
    float* ws    = (float*)d_ws;
    float* hbuf0 = ws;                               // N*128
    float* hbuf1 = hbuf0 + (size_t)GN * GD;          // N*128
    float* agg   = hbuf1 + (size_t)GN * GD;          // N*128
    float* cnt   = agg + (size_t)GN * GD;            // N

    const int aggrs[6] = {AG_SUM, AG_MAX, AG_MEAN, AG_SUM, AG_MAX, AG_MEAN};

    const long long initN = (long long)GN * GD + GN;
    const int initBlocks  = (int)((initN + 255) / 256);
    const int scatBlocks  = (int)(((long long)GE * 32 + 255) / 256);
    const int finBlocks   = (int)(((long long)GN * GD + 255) / 256);
    const int gemmBlocks  = GN / 16;                 // 3125, exact
    const int headBlocks  = (int)(((long long)GN * 32 + 255) / 256);

    const float* h = x;
    float* hb[2] = {hbuf0, hbuf1};

    for (int i = 0; i < 6; ++i) {
        const int aggr = aggrs[i];
        const float fill = (aggr == AG_MAX) ? -__builtin_inff() : 0.0f;

        sage_init<<<initBlocks, 256, 0, stream>>>(agg, cnt, fill);
        sage_scatter<<<scatBlocks, 256, 0, stream>>>(h, ei, agg, cnt, aggr);
        if (aggr != AG_SUM)
            sage_finalize<<<finBlocks, 256, 0, stream>>>(agg, cnt, aggr);

        float* hn = hb[i & 1];
        sage_gemm_wmma<true><<<gemmBlocks, 256, 0, stream>>>(
            agg, w_l + (size_t)i * GD * GD, b_l + (size_t)i * GD,
            h, w_r + (size_t)i * GD * GD, hn);
        h = hn;
    }

    // MLP head: relu(h @ w1 + b1) -> hbuf0 (h currently == hbuf1)
    sage_gemm_wmma<false><<<gemmBlocks, 256, 0, stream>>>(
        h, w1, b1, nullptr, nullptr, hbuf0);

    sage_head<<<headBlocks, 256, 0, stream>>>(hbuf0, w2, b2, out);
}